// RandScatterRouter_6777458393947
// MI455X (gfx1250) — compile-verified
//
#include <hip/hip_runtime.h>
#include <hip/hip_bf16.h>
#include <stdint.h>

// ---------------- problem constants ----------------
#define N_TOK   131072
#define D_MODEL 512
#define P_NUM   16
#define CAP     16384              // 2 * N / P  (= 2^14)
#define TOK_PER_BLOCK 1024
#define NB      (N_TOK / TOK_PER_BLOCK)   // 128
#define CHUNKS  (TOK_PER_BLOCK / 256)     // 4

// ---------------- gfx1250 async global<->LDS path (guarded) ----------------
#if defined(__HIP_DEVICE_COMPILE__) && \
    __has_builtin(__builtin_amdgcn_global_load_async_to_lds_b128) && \
    __has_builtin(__builtin_amdgcn_global_store_async_from_lds_b128) && \
    __has_builtin(__builtin_amdgcn_s_wait_asynccnt)
#define HAVE_ASYNC_LDS 1
typedef int v4i __attribute__((ext_vector_type(4)));
typedef __attribute__((address_space(1))) v4i glb_v4i;   // global (AS1)
typedef __attribute__((address_space(3))) v4i lds_v4i;   // LDS    (AS3)
#else
#define HAVE_ASYNC_LDS 0
#endif

// =====================================================================
// Kernel A: argmax route + stable in-block rank + per-block histogram
// =====================================================================
__global__ __launch_bounds__(256)
void routeKernel(const float* __restrict__ score,
                 uint32_t* __restrict__ packed,      // [N]  (path<<16)|localRank
                 uint32_t* __restrict__ blockCounts) // [NB][16]
{
    const int tid  = threadIdx.x;
    const int lane = tid & 31;
    const int wave = tid >> 5;
    const int blockStart = blockIdx.x * TOK_PER_BLOCK;

    __shared__ uint32_t runCnt[P_NUM];
    __shared__ uint32_t waveCnt[8][P_NUM];

    if (tid < P_NUM) runCnt[tid] = 0;
    __syncthreads();

    for (int c = 0; c < CHUNKS; ++c) {
        const int token = blockStart + c * 256 + tid;

        // load 16 scores (contiguous per token), first-max-wins argmax
        const float4* srow = (const float4*)(score + (size_t)token * P_NUM);
        float4 s0 = srow[0], s1 = srow[1], s2 = srow[2], s3 = srow[3];
        float v[16] = { s0.x, s0.y, s0.z, s0.w,
                        s1.x, s1.y, s1.z, s1.w,
                        s2.x, s2.y, s2.z, s2.w,
                        s3.x, s3.y, s3.z, s3.w };
        int p = 0;
        float best = v[0];
        #pragma unroll
        for (int i = 1; i < 16; ++i) {
            if (v[i] > best) { best = v[i]; p = i; }
        }

        // wave32 per-path ballots: stable rank within wave + per-wave counts
        unsigned myMask = 0u;
        #pragma unroll
        for (int pp = 0; pp < P_NUM; ++pp) {
            unsigned bal = (unsigned)__ballot(p == pp);
            if (p == pp)    myMask = bal;
            if (lane == pp) waveCnt[wave][pp] = (uint32_t)__popc(bal);
        }
        const unsigned laneRank = (unsigned)__popc(myMask & ((1u << lane) - 1u));

        __syncthreads();
        unsigned off = runCnt[p];
        #pragma unroll
        for (int w = 0; w < 8; ++w) {
            if (w < wave) off += waveCnt[w][p];
        }
        const unsigned localRank = off + laneRank;     // < 1024
        packed[token] = ((unsigned)p << 16) | localRank;

        __syncthreads();
        if (tid < P_NUM) {
            unsigned t = 0;
            #pragma unroll
            for (int w = 0; w < 8; ++w) t += waveCnt[w][tid];
            runCnt[tid] += t;
        }
        __syncthreads();
    }

    if (tid < P_NUM) blockCounts[blockIdx.x * P_NUM + tid] = runCnt[tid];
}

// =====================================================================
// Kernel B: exclusive scan of block histograms per path + path totals
// =====================================================================
__global__ void scanKernel(const uint32_t* __restrict__ blockCounts,
                           uint32_t* __restrict__ blockOffsets,
                           uint32_t* __restrict__ pathTotal)
{
    const int t = threadIdx.x;
    if (t < P_NUM) {
        uint32_t run = 0;
        for (int b = 0; b < NB; ++b) {
            uint32_t c = blockCounts[b * P_NUM + t];
            blockOffsets[b * P_NUM + t] = run;
            run += c;
        }
        pathTotal[t] = run;
    }
}

// =====================================================================
// Kernel C: scatter token index into its (path, pos) slot.
// Slots are dense per path (0..count-1), so no slot-map init is needed:
// the gather only reads slots s < min(pathTotal, CAP), all freshly written.
// =====================================================================
__global__ __launch_bounds__(256)
void scatterKernel(const uint32_t* __restrict__ packed,
                   const uint32_t* __restrict__ blockOffsets,
                   int32_t* __restrict__ slotMap)
{
    const int tid = threadIdx.x;
    const int b   = blockIdx.x;
    const int blockStart = b * TOK_PER_BLOCK;
    for (int c = 0; c < CHUNKS; ++c) {
        const int token = blockStart + c * 256 + tid;
        const uint32_t pk = packed[token];
        const unsigned p  = pk >> 16;
        const unsigned r  = pk & 0xFFFFu;
        const unsigned pos = blockOffsets[b * P_NUM + p] + r;
        if (pos < CAP) slotMap[p * CAP + pos] = token;
    }
}

// =====================================================================
// Kernel D: gather — write every output row exactly once (copy or zeros).
// One wave per 2 rows. Async path: issue both rows' global->LDS loads,
// write zero rows while they're in flight, one ASYNCcnt wait, then both
// rows' LDS->global stores.
// =====================================================================
__global__ __launch_bounds__(256)
void gatherKernel(const float* __restrict__ inp,
                  const int32_t* __restrict__ slotMap,
                  const uint32_t* __restrict__ pathTotal,
                  float* __restrict__ out)
{
    const int lane = threadIdx.x & 31;
    const int wave = threadIdx.x >> 5;
    const int rowBase = blockIdx.x * 16 + wave * 2;   // 8 waves * 2 rows

#if HAVE_ASYNC_LDS
    __shared__ float4 stage[8][2][128];               // 32 KB
#endif

    int toks[2];
    #pragma unroll
    for (int r = 0; r < 2; ++r) {
        const int row = rowBase + r;
        const int p = row >> 14;                      // CAP = 2^14
        const int s = row & (CAP - 1);
        toks[r] = (s < (int)pathTotal[p]) ? slotMap[row] : -1;
    }

#if HAVE_ASYNC_LDS
    // 1) issue async global->LDS loads for both occupied rows
    #pragma unroll
    for (int r = 0; r < 2; ++r) {
        if (toks[r] >= 0) {
            const float4* src = (const float4*)(inp + (size_t)toks[r] * D_MODEL);
            float4* st = &stage[wave][r][0];
            #pragma unroll
            for (int k = 0; k < 4; ++k) {
                __builtin_amdgcn_global_load_async_to_lds_b128(
                    (glb_v4i*)&src[lane + 32 * k],
                    (lds_v4i*)&st[lane + 32 * k], 0, 0);
            }
        }
    }
    // 2) zero rows written directly while loads are in flight
    #pragma unroll
    for (int r = 0; r < 2; ++r) {
        if (toks[r] < 0) {
            float4* dst = (float4*)(out + (size_t)(rowBase + r) * D_MODEL);
            const float4 z = {0.0f, 0.0f, 0.0f, 0.0f};
            #pragma unroll
            for (int k = 0; k < 4; ++k) dst[lane + 32 * k] = z;
        }
    }
    // 3) single wait, then async LDS->global stores for both rows
    __builtin_amdgcn_s_wait_asynccnt(0);
    #pragma unroll
    for (int r = 0; r < 2; ++r) {
        if (toks[r] >= 0) {
            float4* dst = (float4*)(out + (size_t)(rowBase + r) * D_MODEL);
            float4* st = &stage[wave][r][0];
            #pragma unroll
            for (int k = 0; k < 4; ++k) {
                __builtin_amdgcn_global_store_async_from_lds_b128(
                    (glb_v4i*)&dst[lane + 32 * k],
                    (lds_v4i*)&st[lane + 32 * k], 0, 0);
            }
        }
    }
    __builtin_amdgcn_s_wait_asynccnt(0);
#else
    #pragma unroll
    for (int r = 0; r < 2; ++r) {
        if (toks[r] >= 0) {
            __builtin_prefetch(inp + (size_t)toks[r] * D_MODEL + lane * 16, 0, 3);
        }
    }
    #pragma unroll
    for (int r = 0; r < 2; ++r) {
        const int tok = toks[r];
        float4* dst = (float4*)(out + (size_t)(rowBase + r) * D_MODEL);
        if (tok >= 0) {
            const float4* src = (const float4*)(inp + (size_t)tok * D_MODEL);
            #pragma unroll
            for (int k = 0; k < 4; ++k) dst[lane + 32 * k] = src[lane + 32 * k];
        } else {
            const float4 z = {0.0f, 0.0f, 0.0f, 0.0f};
            #pragma unroll
            for (int k = 0; k < 4; ++k) dst[lane + 32 * k] = z;
        }
    }
#endif
}

// =====================================================================
// Launcher
// =====================================================================
extern "C" void kernel_launch(void* const* d_in, const int* in_sizes, int n_in,
                              void* d_out, int out_size, void* d_ws, size_t ws_size,
                              hipStream_t stream)
{
    const float* inp   = (const float*)d_in[0];   // [N, 512] f32
    const float* score = (const float*)d_in[1];   // [N, 16]  f32
    float* out = (float*)d_out;                   // [16, 16384, 512] f32

    // workspace layout (~1.6 MB)
    uint32_t* packed       = (uint32_t*)d_ws;                 // N
    uint32_t* blockCounts  = packed + N_TOK;                  // NB*16
    uint32_t* blockOffsets = blockCounts + NB * P_NUM;        // NB*16
    int32_t*  slotMap      = (int32_t*)(blockOffsets + NB * P_NUM); // P*CAP
    uint32_t* pathTotal    = (uint32_t*)(slotMap + P_NUM * CAP);    // 16

    routeKernel  <<<NB, 256, 0, stream>>>(score, packed, blockCounts);
    scanKernel   <<<1, 32, 0, stream>>>(blockCounts, blockOffsets, pathTotal);
    scatterKernel<<<NB, 256, 0, stream>>>(packed, blockOffsets, slotMap);
    gatherKernel <<<(P_NUM * CAP) / 16, 256, 0, stream>>>(inp, slotMap, pathTotal, out);
}